// PostProcess_31181462569586
// MI455X (gfx1250) — compile-verified
//
#include <hip/hip_runtime.h>
#include <stdint.h>

// Problem constants (fixed by the reference).
#define B      64
#define Q      1000
#define C      1203
#define QC     (Q * C)          // 1,203,000 elements per batch
#define TOPK   100
#define KSEL   1000             // k = max(topk, Q)
#define NBINS  8192             // 13-bit histogram of monotone(sigmoid) bits
#define CAP    4096             // candidate capacity per batch (expect ~2.5K)
#define HIST_BPB 32             // blocks per batch, histogram pass
#define COMP_BPB 64             // blocks per batch, compaction pass
#define TILE   1024             // floats per staged tile (4 KB)
#define NTILES ((QC + TILE - 1) / TILE)   // 1175
#define FULLT  (QC / TILE)                // 1174 full tiles, tail = 824 (=4*206)

// Output layout (floats), concatenated in reference return order.
#define OFF_SCORES  0
#define OFF_LABELS  (B * TOPK)                    // 6,400
#define OFF_BOXES   (2 * B * TOPK)                // 12,800
#define OFF_PROB    (2 * B * TOPK + B * TOPK * 4) // 38,400
#define OFF_IDXALL  (OFF_PROB + B * TOPK * C)     // 7,737,600
#define OFF_BOXNORM (OFF_IDXALL + B * Q)          // 7,801,600

__device__ __forceinline__ float sigmoidf_(float x) {
  return 1.0f / (1.0f + expf(-x));
}

// Monotone bit mapping: larger float -> larger uint32.
__device__ __forceinline__ uint32_t fmono(float f) {
  uint32_t u = __float_as_uint(f);
  return (u & 0x80000000u) ? ~u : (u | 0x80000000u);
}

// ---- CDNA5 async global->LDS staging (GVS mode: sgpr base + vgpr byte offset)
__device__ __forceinline__ void async_b128(uint32_t lds_off, uint32_t gbyte_off,
                                           const void* sbase) {
  asm volatile("global_load_async_to_lds_b128 %0, %1, %2"
               :: "v"(lds_off), "v"(gbyte_off), "s"(sbase)
               : "memory");
}
__device__ __forceinline__ void wait_async_le0() {
  asm volatile("s_wait_asynccnt 0" ::: "memory");
}
__device__ __forceinline__ void wait_async_le1() {
  asm volatile("s_wait_asynccnt 1" ::: "memory");
}

// ---------------------------------------------------------------- kernels ----

__global__ __launch_bounds__(256) void initKernel(uint32_t* hist, uint32_t* candCount) {
  int i = blockIdx.x * 256 + threadIdx.x;
  if (i < B * NBINS) hist[i] = 0u;
  else if (i < B * NBINS + B) candCount[i - B * NBINS] = 0u;
}

// Pass 1: per-batch histogram of fmono(sigmoid(logit)) >> 19, streaming logits
// through LDS with double-buffered async loads. Each lane stages and consumes
// its own 16B slot, so only the per-wave ASYNCcnt wait is needed (no barrier).
__global__ __launch_bounds__(256) void histKernel(const float* __restrict__ logits,
                                                  uint32_t* __restrict__ hist) {
  __shared__ uint32_t shist[NBINS];
  __shared__ __align__(16) float stage[2][TILE];

  const int tid = threadIdx.x;
  const int b   = blockIdx.x / HIST_BPB;
  const int blk = blockIdx.x % HIST_BPB;

  for (int i = tid; i < NBINS; i += 256) shist[i] = 0u;
  __syncthreads();

  const float* src = logits + (size_t)b * QC;   // wave-uniform SGPR base
  const uint32_t lds0 = (uint32_t)(uintptr_t)&stage[0][0];
  const uint32_t lds1 = (uint32_t)(uintptr_t)&stage[1][0];
  const uint32_t gmax = (uint32_t)(QC * 4 - 16);
  const uint32_t lane16 = (uint32_t)(tid * 16);

  // Every wave issues every tile (addresses clamped in-bounds) so that the
  // per-wave ASYNCcnt bookkeeping is identical across waves.
  int t = blk;
  int buf = 0;
  {
    uint32_t goff = (uint32_t)(t * TILE) * 4u + lane16;
    if (goff > gmax) goff = gmax;
    async_b128(lds0 + lane16, goff, src);
  }
  for (; t < NTILES; t += HIST_BPB) {
    const int tn = t + HIST_BPB;
    if (tn < NTILES) {
      uint32_t goff = (uint32_t)(tn * TILE) * 4u + lane16;
      if (goff > gmax) goff = gmax;
      async_b128(((buf == 0) ? lds1 : lds0) + lane16, goff, src);
      wait_async_le1();   // oldest (current tile) complete
    } else {
      wait_async_le0();
    }
    const int n = (t < FULLT) ? TILE : (QC - t * TILE);
    const float* sb = (buf == 0) ? stage[0] : stage[1];
#pragma unroll
    for (int e = 0; e < 4; ++e) {
      const int idx = tid * 4 + e;
      if (idx < n) {
        const float p = sigmoidf_(sb[idx]);
        atomicAdd(&shist[fmono(p) >> 19], 1u);
      }
    }
    buf ^= 1;
  }

  __syncthreads();
  uint32_t* gh = hist + (size_t)b * NBINS;
  for (int i = tid; i < NBINS; i += 256) {
    const uint32_t v = shist[i];
    if (v) atomicAdd(&gh[i], v);
  }
}

// Pass 2: find threshold bin per batch (cumulative count from top >= KSEL).
__global__ void thresholdKernel(const uint32_t* __restrict__ hist,
                                uint32_t* __restrict__ thrBin) {
  const int b = blockIdx.x * blockDim.x + threadIdx.x;
  if (b >= B) return;
  const uint32_t* h = hist + (size_t)b * NBINS;
  uint32_t acc = 0;
  int bin = NBINS - 1;
  for (; bin > 0; --bin) {
    acc += h[bin];
    if (acc >= KSEL) break;
  }
  thrBin[b] = (uint32_t)bin;
}

// Pass 3: compact candidates (bin >= threshold) as 64-bit sort keys:
// key = (fmono(p) << 32) | ~flat_idx  -> descending key order == top_k order
// (value desc, ties broken by smaller index, matching jax.lax.top_k).
__global__ __launch_bounds__(256) void compactKernel(const float* __restrict__ logits,
                                                     const uint32_t* __restrict__ thrBin,
                                                     uint32_t* __restrict__ candCount,
                                                     uint64_t* __restrict__ candKeys) {
  const int b   = blockIdx.x / COMP_BPB;
  const int blk = blockIdx.x % COMP_BPB;
  const uint32_t thr = thrBin[b];
  const float4* src = (const float4*)(logits + (size_t)b * QC);
  uint32_t* cnt = candCount + b;
  uint64_t* keys = candKeys + (size_t)b * CAP;
  const int NV = QC / 4;   // 300,750 float4s, QC divisible by 4

  for (int v = blk * 256 + threadIdx.x; v < NV; v += COMP_BPB * 256) {
    const float4 f = src[v];
    const float vals[4] = {f.x, f.y, f.z, f.w};
#pragma unroll
    for (int e = 0; e < 4; ++e) {
      const uint32_t mu = fmono(sigmoidf_(vals[e]));
      if ((mu >> 19) >= thr) {
        const uint32_t pos = atomicAdd(cnt, 1u);
        if (pos < CAP) {
          const uint32_t idx = (uint32_t)(v * 4 + e);
          keys[pos] = ((uint64_t)mu << 32) | (uint32_t)(~idx);
        }
      }
    }
  }
}

// Pass 4: exact sorted top-KSEL per batch via bitonic sort of CAP keys in LDS.
__global__ __launch_bounds__(1024) void sortKernel(const uint64_t* __restrict__ candKeys,
                                                   const uint32_t* __restrict__ candCount,
                                                   uint32_t* __restrict__ sortedIdx,
                                                   float* __restrict__ sortedVal) {
  __shared__ uint64_t sk[CAP];    // 32 KB
  const int b = blockIdx.x;
  const int tid = threadIdx.x;
  uint32_t cnt = candCount[b];
  if (cnt > CAP) cnt = CAP;
  const uint64_t* keys = candKeys + (size_t)b * CAP;

  for (int i = tid; i < CAP; i += 1024) sk[i] = (i < (int)cnt) ? keys[i] : 0ull;
  __syncthreads();

  for (int k = 2; k <= CAP; k <<= 1) {
    for (int j = k >> 1; j > 0; j >>= 1) {
      for (int i = tid; i < CAP; i += 1024) {
        const int ixj = i ^ j;
        if (ixj > i) {
          const uint64_t a = sk[i], c2 = sk[ixj];
          const bool up = ((i & k) == 0);
          if (up ? (a < c2) : (a > c2)) { sk[i] = c2; sk[ixj] = a; }  // descending
        }
      }
      __syncthreads();
    }
  }

  for (int i = tid; i < KSEL; i += 1024) {
    const uint64_t key = sk[i];
    const uint32_t mu = (uint32_t)(key >> 32);
    const uint32_t u  = (mu & 0x80000000u) ? (mu & 0x7FFFFFFFu) : ~mu;
    sortedVal[(size_t)b * KSEL + i] = __uint_as_float(u);   // sigmoid prob
    sortedIdx[(size_t)b * KSEL + i] = ~((uint32_t)key);     // flat index
  }
}

// Pass 5: scores, labels, boxes (norm + scaled), topk_indexes_all.
__global__ __launch_bounds__(256) void outputsKernel(const float* __restrict__ boxesIn,
                                                     const float* __restrict__ tsizes,
                                                     const uint32_t* __restrict__ sortedIdx,
                                                     const float* __restrict__ sortedVal,
                                                     float* __restrict__ out) {
  const int b = blockIdx.x;
  const int tid = threadIdx.x;
  const float img_h = tsizes[b * 2 + 0];
  const float img_w = tsizes[b * 2 + 1];
  float* outScores = out + OFF_SCORES  + (size_t)b * TOPK;
  float* outLabels = out + OFF_LABELS  + (size_t)b * TOPK;
  float* outBoxes  = out + OFF_BOXES   + (size_t)b * TOPK * 4;
  float* outBN     = out + OFF_BOXNORM + (size_t)b * TOPK * 4;
  float* outIdxAll = out + OFF_IDXALL  + (size_t)b * Q;

  for (int j = tid; j < TOPK; j += 256) {
    const uint32_t idx = sortedIdx[(size_t)b * KSEL + j];
    outScores[j] = sortedVal[(size_t)b * KSEL + j];
    outLabels[j] = (float)(idx % C);
    const uint32_t q = idx / C;
    const float* bx = boxesIn + ((size_t)b * Q + q) * 4;
    const float cx = bx[0], cy = bx[1], w = bx[2], h = bx[3];
    const float x0 = cx - 0.5f * w, y0 = cy - 0.5f * h;
    const float x1 = cx + 0.5f * w, y1 = cy + 0.5f * h;
    outBN[j * 4 + 0] = x0;         outBN[j * 4 + 1] = y0;
    outBN[j * 4 + 2] = x1;         outBN[j * 4 + 3] = y1;
    outBoxes[j * 4 + 0] = x0 * img_w; outBoxes[j * 4 + 1] = y0 * img_h;
    outBoxes[j * 4 + 2] = x1 * img_w; outBoxes[j * 4 + 3] = y1 * img_h;
  }
  for (int i = tid; i < Q; i += 256) {
    outIdxAll[i] = (float)(sortedIdx[(size_t)b * KSEL + i] / C);
  }
}

// Pass 6: topk_inst_prob = sigmoid(pred_logits[b, q_j, :]) gathered per top row.
__global__ __launch_bounds__(256) void gatherProbKernel(const float* __restrict__ logits,
                                                        const uint32_t* __restrict__ sortedIdx,
                                                        float* __restrict__ outProb) {
  const int b = blockIdx.x / TOPK;
  const int j = blockIdx.x % TOPK;
  const uint32_t q = sortedIdx[(size_t)b * KSEL + j] / C;
  const float* src = logits + ((size_t)b * Q + q) * C;
  float* dst = outProb + (size_t)blockIdx.x * C;
  for (int c = threadIdx.x; c < C; c += 256) dst[c] = sigmoidf_(src[c]);
}

// ---------------------------------------------------------------- launch -----

extern "C" void kernel_launch(void* const* d_in, const int* in_sizes, int n_in,
                              void* d_out, int out_size, void* d_ws, size_t ws_size,
                              hipStream_t stream) {
  const float* logits = (const float*)d_in[0];   // (B,Q,C) f32
  const float* boxes  = (const float*)d_in[1];   // (B,Q,4) f32
  const float* tsz    = (const float*)d_in[2];   // (B,2)   f32
  (void)in_sizes; (void)n_in; (void)out_size; (void)ws_size;

  char* ws = (char*)d_ws;
  size_t off = 0;
  uint32_t* hist      = (uint32_t*)(ws + off); off += (size_t)B * NBINS * 4;  // 2 MB
  uint32_t* thrBin    = (uint32_t*)(ws + off); off += (size_t)B * 4;
  uint32_t* candCount = (uint32_t*)(ws + off); off += (size_t)B * 4;
  uint64_t* candKeys  = (uint64_t*)(ws + off); off += (size_t)B * CAP * 8;    // 2 MB
  uint32_t* sortedIdx = (uint32_t*)(ws + off); off += (size_t)B * KSEL * 4;
  float*    sortedVal = (float*)(ws + off);    off += (size_t)B * KSEL * 4;
  float* out = (float*)d_out;

  const int initN = B * NBINS + B;
  initKernel<<<(initN + 255) / 256, 256, 0, stream>>>(hist, candCount);
  histKernel<<<B * HIST_BPB, 256, 0, stream>>>(logits, hist);
  thresholdKernel<<<1, 64, 0, stream>>>(hist, thrBin);
  compactKernel<<<B * COMP_BPB, 256, 0, stream>>>(logits, thrBin, candCount, candKeys);
  sortKernel<<<B, 1024, 0, stream>>>(candKeys, candCount, sortedIdx, sortedVal);
  outputsKernel<<<B, 256, 0, stream>>>(boxes, tsz, sortedIdx, sortedVal, out);
  gatherProbKernel<<<B * TOPK, 256, 0, stream>>>(logits, sortedIdx, out + OFF_PROB);
}